// GMMGate_36421322670722
// MI455X (gfx1250) — compile-verified
//
#include <hip/hip_runtime.h>
#include <hip/hip_bf16.h>

#define MODEL_DIM 4096
#define PROJ_DIM 4
#define TOTAL_COMP 64
#define NUM_EXPERTS 16
#define BTOK 8192
#define CAPACITY 1024
#define LOG2PI_F 1.8378770664093453f

typedef __attribute__((ext_vector_type(2))) float v2f;
typedef __attribute__((ext_vector_type(4))) float v4f;
typedef __attribute__((ext_vector_type(8))) float v8f;

// ---- d_out layout (outputs concatenated flat, all f32) ----
#define O_LOSS     0ull
#define O_COMBINE  1ull
#define N_COMBINE  (8192ull * 16ull * 1024ull)          // 134217728
#define O_DISPATCH (O_COMBINE + N_COMBINE)              // 134217729
#define O_EXPCNT   (O_DISPATCH + N_COMBINE)             // 268435457
#define O_GATES    (O_EXPCNT + 16ull)                   // 268435473

// ---- workspace layout (bytes) ----
#define WS_PROJ  0          // float[8192*4]
#define WS_G1    131072     // float[8192]
#define WS_G2    163840     // float[8192]
#define WS_IDX1  196608     // int[8192]
#define WS_IDX2  229376     // int[8192]
#define WS_LOC1  262144     // int[8192]
#define WS_LOC2  294912     // int[8192]
#define WS_CNT1  327680     // int[16]
#define WS_CNT2  327744     // int[16]
#define WS_ACC   327808     // float[2]: [0]=sum(lse), [1]=sum(sqerr)

// ------------------------------------------------------------------
// init: zero the two float accumulators
// ------------------------------------------------------------------
__global__ void init_kernel(float* accum) {
    if (threadIdx.x < 2) accum[threadIdx.x] = 0.0f;
}

// ------------------------------------------------------------------
// Kernel 1: fused down-projection + reconstruction loss.
// One block (8 waves) per 16-row tile. WMMA f32 16x16x4 for both GEMMs.
// ------------------------------------------------------------------
__global__ __launch_bounds__(256) void proj_recon_kernel(
    const float* __restrict__ x, const float* __restrict__ Wd,
    const float* __restrict__ Wu, float* __restrict__ proj_out,
    float* __restrict__ accum)
{
    __shared__ float sproj[16 * PROJ_DIM];
    __shared__ float sacc[8];
    const int tid  = threadIdx.x;
    const int wave = tid >> 5;
    const int lane = tid & 31;
    const int hi   = lane >> 4;          // second half-wave
    const int col  = lane & 15;
    const long rowBase = (long)blockIdx.x * 16;

    if (tid < 16 * PROJ_DIM) sproj[tid] = 0.0f;
    __syncthreads();

    // ---- phase 1: proj(16x4) = x(16x4096) @ W_down(4096x4), K split across waves ----
    {
        v8f acc = {};
        const float* xrow = x + (rowBase + col) * (long)MODEL_DIM;  // M = lane&15
        const int kbeg = wave * 512;
        for (int k0 = kbeg; k0 < kbeg + 512; k0 += 4) {
            const int ka = k0 + (hi ? 2 : 0);
            __builtin_prefetch(xrow + ka + 256, 0, 0);     // global_prefetch
            v2f a; a.x = xrow[ka]; a.y = xrow[ka + 1];     // A 16x4 layout (ISA 7.12.2)
            v2f b;
            if (col < PROJ_DIM) {                          // B 4x16, cols 4..15 zero-padded
                b.x = Wd[(k0 + (hi ? 2 : 0)) * PROJ_DIM + col];
                b.y = Wd[(k0 + (hi ? 3 : 1)) * PROJ_DIM + col];
            } else { b.x = 0.0f; b.y = 0.0f; }
            acc = __builtin_amdgcn_wmma_f32_16x16x4_f32(
                      false, a, false, b, (short)0, acc, false, false);
        }
        if (col < PROJ_DIM) {                              // merge wave partials (N<4 valid)
            #pragma unroll
            for (int r = 0; r < 8; ++r) {
                const int m = r + (hi ? 8 : 0);            // D layout: vgpr r -> row r / r+8
                atomicAdd(&sproj[m * PROJ_DIM + col], acc[r]);
            }
        }
    }
    __syncthreads();
    if (tid < 16 * PROJ_DIM) proj_out[rowBase * PROJ_DIM + tid] = sproj[tid];

    // ---- phase 2: recon = proj @ W_up, accumulate (recon - x)^2 ----
    float sq = 0.0f;
    {
        v2f a;
        { const int m = lane & 15;                         // A = proj tile from LDS
          a.x = sproj[m * PROJ_DIM + (hi ? 2 : 0)];
          a.y = sproj[m * PROJ_DIM + (hi ? 3 : 1)]; }
        for (int c = 0; c < 32; ++c) {                     // 256 N-chunks / 8 waves
            const int n0 = (wave * 32 + c) * 16;
            v2f b;
            b.x = Wu[(hi ? 2 : 0) * MODEL_DIM + n0 + col];
            b.y = Wu[(hi ? 3 : 1) * MODEL_DIM + n0 + col];
            v8f z = {};
            v8f d = __builtin_amdgcn_wmma_f32_16x16x4_f32(
                        false, a, false, b, (short)0, z, false, false);
            #pragma unroll
            for (int r = 0; r < 8; ++r) {
                const long m = rowBase + r + (hi ? 8 : 0);
                const float xv = x[m * (long)MODEL_DIM + n0 + col];  // L2-resident re-read
                const float df = d[r] - xv;
                sq += df * df;
            }
        }
    }
    for (int off = 16; off > 0; off >>= 1) sq += __shfl_xor(sq, off, 32);
    if (lane == 0) sacc[wave] = sq;
    __syncthreads();
    if (tid == 0) {
        float t = 0.0f;
        #pragma unroll
        for (int w = 0; w < 8; ++w) t += sacc[w];
        atomicAdd(&accum[1], t);
    }
}

// ------------------------------------------------------------------
// Kernel 2: per-token GMM posterior, NLL, expert logits, softmax gates,
// top-1 / top-2 expert selection.
// ------------------------------------------------------------------
__global__ __launch_bounds__(256) void gmm_gate_kernel(
    const float* __restrict__ proj, const unsigned char* __restrict__ drop,
    const float* __restrict__ means, const float* __restrict__ logvars,
    const float* __restrict__ mix, float* __restrict__ gates_out,
    int* __restrict__ idx1_out, int* __restrict__ idx2_out,
    float* __restrict__ g1_out, float* __restrict__ g2_out,
    float* __restrict__ accum)
{
    const int s = blockIdx.x * blockDim.x + threadIdx.x;
    if (s >= BTOK) return;
    const float z0 = proj[s*4+0], z1 = proj[s*4+1], z2 = proj[s*4+2], z3 = proj[s*4+3];

    // log_softmax(mix_logits): streaming logsumexp over 64
    float mm = -__builtin_inff(), ms = 0.0f;
    for (int c = 0; c < TOTAL_COMP; ++c) {
        const float v = mix[c];
        if (v > mm) { ms = ms * __expf(mm - v) + 1.0f; mm = v; }
        else        { ms += __expf(v - mm); }
    }
    const float mixLse = mm + __logf(ms);

    auto clogit = [&](int c) -> float {
        float md = 0.0f, ld = 0.0f;
        #pragma unroll
        for (int p = 0; p < PROJ_DIM; ++p) {
            const float mu = means[c*PROJ_DIM + p];
            const float lv = logvars[c*PROJ_DIM + p];
            const float zz = (p == 0 ? z0 : p == 1 ? z1 : p == 2 ? z2 : z3);
            const float dd = zz - mu;
            md += dd * dd / (__expf(lv) + 1e-6f);
            ld += lv;
        }
        return -0.5f * (md + ld + (float)PROJ_DIM * LOG2PI_F) + (mix[c] - mixLse);
    };

    // streaming logsumexp of gmm logits
    float gm = -__builtin_inff(), gs = 0.0f;
    for (int c = 0; c < TOTAL_COMP; ++c) {
        if (drop[c]) continue;
        const float l = clogit(c);
        if (l > gm) { gs = gs * __expf(gm - l) + 1.0f; gm = l; }
        else        { gs += __expf(l - gm); }
    }
    const float lse = gm + __logf(gs);

    // accumulate NLL term (wave-reduced)
    float lsum = lse;
    for (int off = 16; off > 0; off >>= 1) lsum += __shfl_xor(lsum, off, 32);
    if ((threadIdx.x & 31) == 0) atomicAdd(&accum[0], lsum);

    // expert posterior = max over 4 components
    float ep[NUM_EXPERTS];
    for (int e = 0; e < NUM_EXPERTS; ++e) {
        float best = 0.0f;
        #pragma unroll
        for (int j = 0; j < 4; ++j) {
            const int c = e*4 + j;
            if (drop[c]) continue;
            best = fmaxf(best, __expf(clogit(c) - lse));
        }
        ep[e] = best;
    }

    // expert_logits = ep + roll(ep, 1); gates = softmax
    float el[NUM_EXPERTS], g[NUM_EXPERTS];
    float mx = -__builtin_inff();
    #pragma unroll
    for (int e = 0; e < NUM_EXPERTS; ++e) {
        el[e] = ep[e] + ep[(e + 15) & 15];
        mx = fmaxf(mx, el[e]);
    }
    float sum = 0.0f;
    #pragma unroll
    for (int e = 0; e < NUM_EXPERTS; ++e) { g[e] = __expf(el[e] - mx); sum += g[e]; }
    const float inv = 1.0f / sum;
    int bi = 0; float bg = -1.0f;
    #pragma unroll
    for (int e = 0; e < NUM_EXPERTS; ++e) {
        g[e] *= inv;
        gates_out[(long)s * NUM_EXPERTS + e] = g[e];
        if (g[e] > bg) { bg = g[e]; bi = e; }           // first-index ties like argmax
    }
    int bi2 = 0; float bl = -__builtin_inff();
    #pragma unroll
    for (int e = 0; e < NUM_EXPERTS; ++e)
        if (e != bi && el[e] > bl) { bl = el[e]; bi2 = e; }

    idx1_out[s] = bi;  g1_out[s] = g[bi];
    idx2_out[s] = bi2; g2_out[s] = g[bi2];
}

// ------------------------------------------------------------------
// Kernel 3: two-level prefix scan of one-hot masks along token axis.
// One 1024-thread block: (which, chunk 0..31, expert 0..15).
// ------------------------------------------------------------------
__global__ __launch_bounds__(1024) void scan_kernel(
    const int* __restrict__ idx1, const int* __restrict__ idx2,
    int* __restrict__ loc1, int* __restrict__ loc2raw,
    int* __restrict__ cnt1g, int* __restrict__ cnt2g,
    float* __restrict__ expcnt_out)
{
    __shared__ int cnt[2][32][NUM_EXPERTS];
    __shared__ int tot[2][NUM_EXPERTS];
    const int t     = threadIdx.x;
    const int which = t >> 9;
    const int chunk = (t >> 4) & 31;
    const int e     = t & 15;
    const int* idx  = which ? idx2 : idx1;
    const int s0    = chunk * 256;

    int c = 0;
    for (int i = 0; i < 256; ++i) c += (idx[s0 + i] == e);
    cnt[which][chunk][e] = c;
    __syncthreads();
    if (chunk == 0) {
        int run = 0;
        for (int k = 0; k < 32; ++k) { const int v = cnt[which][k][e]; cnt[which][k][e] = run; run += v; }
        tot[which][e] = run;
        if (which == 0) cnt1g[e] = run; else cnt2g[e] = run;
    }
    __syncthreads();
    int off = cnt[which][chunk][e];
    int* loc = which ? loc2raw : loc1;
    for (int i = 0; i < 256; ++i) {
        const int s = s0 + i;
        if (idx[s] == e) { loc[s] = off; ++off; }
    }
    if (t < NUM_EXPERTS) expcnt_out[t] = (float)(tot[0][t] + tot[1][t]);
}

// ------------------------------------------------------------------
// Kernel 4: zero-fill combine + dispatch (1 GiB) with wide NT stores.
// ------------------------------------------------------------------
__global__ void zerofill_kernel(float* __restrict__ out)
{
    // region = floats [1, 1 + 2*N_COMBINE); 16B-aligned vectors start at idx 4
    const unsigned long long nvec = (2ull * N_COMBINE - 3ull) / 4ull;  // 67108863
    v4f* p = (v4f*)(out + 4);
    unsigned long long i      = (unsigned long long)blockIdx.x * blockDim.x + threadIdx.x;
    const unsigned long long stride = (unsigned long long)gridDim.x * blockDim.x;
    const v4f z = {0.f, 0.f, 0.f, 0.f};
    for (; i < nvec; i += stride) __builtin_nontemporal_store(z, &p[i]);
    if (blockIdx.x == 0 && threadIdx.x == 0) {
        out[1] = 0.f; out[2] = 0.f; out[3] = 0.f;
        out[O_DISPATCH + N_COMBINE - 1ull] = 0.f;
    }
}

// ------------------------------------------------------------------
// Kernel 5: capacity masking, gate renorm, scatter, final loss.
// ------------------------------------------------------------------
__global__ __launch_bounds__(256) void scatter_kernel(
    const int* __restrict__ idx1, const int* __restrict__ idx2,
    const int* __restrict__ loc1, const int* __restrict__ loc2raw,
    const int* __restrict__ cnt1g,
    const float* __restrict__ g1v, const float* __restrict__ g2v,
    const float* __restrict__ accum, float* __restrict__ out)
{
    const int s = blockIdx.x * blockDim.x + threadIdx.x;
    if (s >= BTOK) return;
    const int e1 = idx1[s], e2 = idx2[s];
    const int l1 = loc1[s];
    const int l2 = loc2raw[s] + cnt1g[e2];
    const bool k1 = (l1 < CAPACITY), k2 = (l2 < CAPACITY);
    float g1 = k1 ? g1v[s] : 0.0f;
    float g2 = k2 ? g2v[s] : 0.0f;
    const float denom = fmaxf(g1 + g2, 1.1920929e-07f);
    g1 /= denom; g2 /= denom;
    float* comb = out + O_COMBINE;
    float* disp = out + O_DISPATCH;
    const unsigned long long base = (unsigned long long)s * (NUM_EXPERTS * CAPACITY);
    if (k1) {
        const unsigned long long o = base + (unsigned long long)e1 * CAPACITY + l1;
        comb[o] = g1; disp[o] = (g1 != 0.0f) ? 1.0f : 0.0f;
    }
    if (k2) {
        const unsigned long long o = base + (unsigned long long)e2 * CAPACITY + l2;
        comb[o] = g2; disp[o] = (g2 != 0.0f) ? 1.0f : 0.0f;
    }
    if (s == 0)
        out[O_LOSS] = -accum[0] / (float)BTOK
                    +  accum[1] / ((float)BTOK * (float)MODEL_DIM);
}

// ------------------------------------------------------------------
extern "C" void kernel_launch(void* const* d_in, const int* in_sizes, int n_in,
                              void* d_out, int out_size, void* d_ws, size_t ws_size,
                              hipStream_t stream) {
    (void)in_sizes; (void)n_in; (void)out_size; (void)ws_size;
    const float*         x       = (const float*)d_in[0];
    const unsigned char* drop    = (const unsigned char*)d_in[1];
    const float*         means   = (const float*)d_in[2];
    const float*         logvars = (const float*)d_in[3];
    const float*         mix     = (const float*)d_in[4];
    const float*         Wd      = (const float*)d_in[5];
    const float*         Wu      = (const float*)d_in[6];
    float* out = (float*)d_out;
    char*  ws  = (char*)d_ws;

    float* proj  = (float*)(ws + WS_PROJ);
    float* g1v   = (float*)(ws + WS_G1);
    float* g2v   = (float*)(ws + WS_G2);
    int*   idx1  = (int*)(ws + WS_IDX1);
    int*   idx2  = (int*)(ws + WS_IDX2);
    int*   loc1  = (int*)(ws + WS_LOC1);
    int*   loc2  = (int*)(ws + WS_LOC2);
    int*   cnt1  = (int*)(ws + WS_CNT1);
    int*   cnt2  = (int*)(ws + WS_CNT2);
    float* accum = (float*)(ws + WS_ACC);

    init_kernel<<<1, 32, 0, stream>>>(accum);
    zerofill_kernel<<<4096, 256, 0, stream>>>(out);
    proj_recon_kernel<<<BTOK / 16, 256, 0, stream>>>(x, Wd, Wu, proj, accum);
    gmm_gate_kernel<<<BTOK / 256, 256, 0, stream>>>(
        proj, drop, means, logvars, mix, out + O_GATES,
        idx1, idx2, g1v, g2v, accum);
    scan_kernel<<<1, 1024, 0, stream>>>(idx1, idx2, loc1, loc2, cnt1, cnt2,
                                        out + O_EXPCNT);
    scatter_kernel<<<BTOK / 256, 256, 0, stream>>>(
        idx1, idx2, loc1, loc2, cnt1, g1v, g2v, accum, out);
}